// LocalContrastiveLoss_10496900071494
// MI455X (gfx1250) — compile-verified
//
#include <hip/hip_runtime.h>
#include <math.h>

// ---------------- problem constants (match reference) ----------------
#define Bsz   8
#define Dd    128
#define Hh    256
#define Ww    256
#define NBp   15
#define HALFp 7
#define NSa   256
#define NPIX  (NBp * NBp)      // 225
#define MP    240              // padded M (15 tiles of 16)
#define STR   (Dd + 4)         // 132-float LDS row stride: bank-friendly, 8B aligned
#define CENTER 112             // 7*15+7
#define TEMP_F 0.07f
#define EPS_F  1e-8f

typedef __attribute__((ext_vector_type(2))) float v2f;
typedef __attribute__((ext_vector_type(8))) float v8f;

__device__ __forceinline__ float waveReduceSum(float v) {
    // wave32 on gfx1250
    for (int off = 16; off > 0; off >>= 1) v += __shfl_xor(v, off, 32);
    return v;
}
__device__ __forceinline__ float waveReduceMax(float v) {
    for (int off = 16; off > 0; off >>= 1) v = fmaxf(v, __shfl_xor(v, off, 32));
    return v;
}

// Low 32 bits of a generic pointer into __shared__ are the wave-relative LDS
// offset (flat->LDS aperture mapping truncates to addr[31:0]), which is what
// the async-to-LDS VDST operand expects.
__device__ __forceinline__ unsigned lds_off_u32(const void* p) {
    return (unsigned)(uintptr_t)p;
}

__global__ __launch_bounds__(256) void lcl_anchor_kernel(
    const float* __restrict__ feat,   // (B, D, H, W)
    const int*   __restrict__ targ,   // (B, H, W)
    const int*   __restrict__ ah_,    // (B, NS)
    const int*   __restrict__ aw_,    // (B, NS)
    float*       __restrict__ ws)     // (B*NS, 2) : loss_i, valid
{
    __shared__ __align__(16) float patch[MP * STR];   // ~124 KB
    __shared__ float pn[MP];
    __shared__ float simv[MP];
    __shared__ float redA[8], redB[8], redC[8];
    __shared__ float gshare;

    const int blk = blockIdx.x;           // 0..2047
    const int b   = blk / NSa;
    const int tid = threadIdx.x;
    const int ah  = ah_[blk];
    const int aw  = aw_[blk];

    const float* fb = feat + (size_t)b * Dd * Hh * Ww;

    // ---- stage patch (225 pixels x 128 dims) into LDS via async-to-LDS ----
    // GLOBAL_LOAD_ASYNC_TO_LDS_B32: per-lane gather, no VGPR round-trip,
    // tracked on ASYNCcnt so all copies stay in flight with zero reg pressure.
    // Consecutive tids -> consecutive px -> coalesced 60B runs along w.
    for (int idx = tid; idx < Dd * NPIX; idx += 256) {
        int d  = idx / NPIX;
        int p  = idx - d * NPIX;
        int py = p / NBp;
        int px = p - py * NBp;
        int hh = ah - HALFp + py;
        int wp = aw - HALFp + px;
        unsigned goff = (unsigned)((((d * Hh) + hh) * Ww + wp) * 4);   // byte offset < 32MB
        unsigned loff = lds_off_u32(&patch[p * STR + d]);              // LDS byte offset
        asm volatile("global_load_async_to_lds_b32 %0, %1, %2 offset:0"
                     :: "v"(loff), "v"(goff), "s"(fb) : "memory");
    }
    // zero the pad rows 225..239 so WMMA on tile 14 stays finite
    for (int idx = tid; idx < (MP - NPIX) * Dd; idx += 256) {
        int d = idx % Dd;
        int p = NPIX + idx / Dd;
        patch[p * STR + d] = 0.f;
    }
    asm volatile("s_wait_asynccnt 0x0" ::: "memory");
    __syncthreads();

    // ---- per-pixel norms (anchor feature == center pixel row 112) ----
    if (tid < NPIX) {
        const float* row = &patch[tid * STR];
        float s = 0.f;
        #pragma unroll 4
        for (int d = 0; d < Dd; d += 2) {
            v2f x = *(const v2f*)(row + d);
            s += x.x * x.x + x.y * x.y;
        }
        pn[tid] = fmaxf(__builtin_sqrtf(s), EPS_F);
    }
    __syncthreads();

    // ---- WMMA pass: D[m][*] = sum_k patch[m][k] * af[k]  (af replicated over N) ----
    const int wave  = tid >> 5;
    const int lane  = tid & 31;
    const int m16   = lane & 15;
    const int khalf = (lane >> 4) * 2;    // K sub-offset per A/B f32 fragment layout
    const float an  = pn[CENTER];

    for (int tile = wave; tile < 15; tile += 8) {   // wave-uniform: EXEC stays full
        v8f acc = {};
        const float* arow = &patch[(tile * 16 + m16) * STR + khalf];
        const float* brow = &patch[CENTER * STR + khalf];
        #pragma unroll
        for (int kk = 0; kk < Dd / 4; ++kk) {
            v2f a  = *(const v2f*)(arow + kk * 4);   // A: 16x4 f32 fragment
            v2f bb = *(const v2f*)(brow + kk * 4);   // B: af broadcast to all 16 cols
            acc = __builtin_amdgcn_wmma_f32_16x16x4_f32(
                    false, a, false, bb, (short)0, acc, false, false);
        }
        if (m16 == 0) {   // lane 0 -> M=r, lane 16 -> M=r+8 (C/D layout)
            int mbase = tile * 16 + (lane >> 4) * 8;
            #pragma unroll
            for (int r = 0; r < 8; ++r) {
                int m = mbase + r;
                if (m < NPIX) simv[m] = acc[r] / (an * pn[m] * TEMP_F);
            }
        }
    }
    __syncthreads();

    // ---- masked logsumexp + positive mean, block reduction ----
    const int alab = targ[((size_t)b * Hh + ah) * Ww + aw];
    float my_sim = -INFINITY, my_psim = 0.f, my_cnt = 0.f;
    if (tid < NPIX) {
        my_sim = simv[tid];
        int py = tid / NBp, px = tid - py * NBp;
        int plab = targ[((size_t)b * Hh + (ah - HALFp + py)) * Ww + (aw - HALFp + px)];
        if (plab == alab && tid != CENTER) { my_psim = my_sim; my_cnt = 1.f; }
    }

    float wm = waveReduceMax(my_sim);
    if (lane == 0) redA[wave] = wm;
    __syncthreads();
    if (tid == 0) {
        float m = redA[0];
        for (int i = 1; i < 8; ++i) m = fmaxf(m, redA[i]);
        gshare = m;
    }
    __syncthreads();
    const float gmax = gshare;

    float e  = (tid < NPIX) ? __expf(my_sim - gmax) : 0.f;
    float se = waveReduceSum(e);
    float sp = waveReduceSum(my_psim);
    float sc = waveReduceSum(my_cnt);
    if (lane == 0) { redA[wave] = se; redB[wave] = sp; redC[wave] = sc; }
    __syncthreads();
    if (tid == 0) {
        float tse = 0.f, tsp = 0.f, tsc = 0.f;
        for (int i = 0; i < 8; ++i) { tse += redA[i]; tsp += redB[i]; tsc += redC[i]; }
        float lse   = gmax + __logf(tse);
        float loss  = 0.f, valid = 0.f;
        if (tsc > 0.f) {
            loss  = -(tsp / fmaxf(tsc, 1.f) - lse);
            valid = 1.f;
        }
        ws[2 * blk + 0] = loss;
        ws[2 * blk + 1] = valid;
    }
}

__global__ __launch_bounds__(256) void lcl_finalize_kernel(
    const float* __restrict__ ws, float* __restrict__ out)
{
    __shared__ float sl[256], sv[256];
    const int tid = threadIdx.x;
    float l = 0.f, v = 0.f;
    for (int i = tid; i < Bsz * NSa; i += 256) {
        l += ws[2 * i + 0];
        v += ws[2 * i + 1];
    }
    sl[tid] = l; sv[tid] = v;
    __syncthreads();
    for (int s = 128; s > 0; s >>= 1) {
        if (tid < s) { sl[tid] += sl[tid + s]; sv[tid] += sv[tid + s]; }
        __syncthreads();
    }
    if (tid == 0)
        out[0] = (sv[0] > 0.f) ? sl[0] / fmaxf(sv[0], 1.f) : 0.f;
}

extern "C" void kernel_launch(void* const* d_in, const int* in_sizes, int n_in,
                              void* d_out, int out_size, void* d_ws, size_t ws_size,
                              hipStream_t stream) {
    const float* feat = (const float*)d_in[0];
    const int*   targ = (const int*)d_in[1];
    const int*   ah   = (const int*)d_in[2];
    const int*   aw   = (const int*)d_in[3];
    float* ws  = (float*)d_ws;
    float* out = (float*)d_out;

    lcl_anchor_kernel<<<Bsz * NSa, 256, 0, stream>>>(feat, targ, ah, aw, ws);
    lcl_finalize_kernel<<<1, 256, 0, stream>>>(ws, out);
}